// SpectralGCN_4389456577462
// MI455X (gfx1250) — compile-verified
//
#include <hip/hip_runtime.h>
#include <hip/hip_bf16.h>

typedef __attribute__((ext_vector_type(2))) float v2f;
typedef __attribute__((ext_vector_type(8))) float v8f;

// ---------------------------------------------------------------------------
// Degree / normalization kernels
// ---------------------------------------------------------------------------
__global__ void gcn_fill_kernel(float* __restrict__ p, float v, int n) {
    int i = blockIdx.x * blockDim.x + threadIdx.x;
    if (i < n) p[i] = v;
}

__global__ void gcn_deg_accum_kernel(const int* __restrict__ dst,
                                     float* __restrict__ deg, int e) {
    int i = blockIdx.x * blockDim.x + threadIdx.x;
    if (i < e) atomicAdd(&deg[dst[i]], 1.0f);
}

__global__ void gcn_rsqrt_kernel(float* __restrict__ p, int n) {
    int i = blockIdx.x * blockDim.x + threadIdx.x;
    if (i < n) p[i] = rsqrtf(p[i]);
}

__global__ void gcn_relu_kernel(const float* __restrict__ x,
                                float* __restrict__ y, int n) {
    int i = blockIdx.x * blockDim.x + threadIdx.x;
    if (i < n) y[i] = fmaxf(x[i], 0.0f);
}

// ---------------------------------------------------------------------------
// WMMA f32 GEMM:  h = X @ W   (X: [N,K], W: [K,D], row-major, 16|N, 16|D, 4|K)
// Also initializes agg = dinv^2 * h + bias  (self-loop + bias term fused).
// One wave32 per 16x16 output tile, V_WMMA_F32_16X16X4_F32 over K.
// ---------------------------------------------------------------------------
__global__ void gcn_gemm_wmma_kernel(const float* __restrict__ X,
                                     const float* __restrict__ W,
                                     const float* __restrict__ bias,
                                     const float* __restrict__ dinv,
                                     float* __restrict__ h,
                                     float* __restrict__ agg,
                                     int N, int K, int D) {
    const int lane  = threadIdx.x & 31;
    const int wib   = threadIdx.x >> 5;                       // wave in block
    const int tile  = blockIdx.x * (blockDim.x >> 5) + wib;
    const int tilesN = D >> 4;                                // col tiles
    const int tilesM = N >> 4;                                // row tiles
    const int tm = tile / tilesN;
    const int tn = tile - tm * tilesN;
    if (tm >= tilesM) return;                                 // wave-uniform guard

    const int  hl = lane & 15;
    const bool hi = lane >= 16;

    // A: lanes 0-15 -> K = k+0/k+1 of row M=hl ; lanes 16-31 -> K = k+2/k+3
    const float* __restrict__ xrow =
        X + (long long)((tm << 4) + hl) * K + (hi ? 2 : 0);
    // B: lanes 0-15 -> rows k+0/k+1, col = tn*16+hl ; lanes 16-31 -> rows k+2/k+3
    const float* __restrict__ wcol =
        W + ((tn << 4) + hl) + (hi ? (long long)2 * D : 0ll);

    v8f c = {};
    for (int k = 0; k < K; k += 4) {
        v2f a, b;
        a.x = xrow[k];
        a.y = xrow[k + 1];
        b.x = wcol[(long long)k * D];
        b.y = wcol[(long long)(k + 1) * D];
        // D = A*B + C  : v_wmma_f32_16x16x4_f32
        c = __builtin_amdgcn_wmma_f32_16x16x4_f32(
                /*neg_a=*/false, a, /*neg_b=*/false, b,
                /*c_mod=*/(short)0, c, /*reuse_a=*/false, /*reuse_b=*/false);
    }

    // C/D layout: VGPR v -> M = tm*16 + v (+8 for hi lanes), N = tn*16 + hl
    const int col   = (tn << 4) + hl;
    const float bc  = bias[col];
    const int rbase = (tm << 4) + (hi ? 8 : 0);
#pragma unroll
    for (int v = 0; v < 8; ++v) {
        const int r = rbase + v;
        const float val = c[v];
        const long long off = (long long)r * D + col;
        h[off] = val;
        const float di = dinv[r];
        agg[off] = di * di * val + bc;                        // self-loop + bias
    }
}

// ---------------------------------------------------------------------------
// Edge aggregation: agg[dst] += dinv[src]*dinv[dst] * h[src]
// One wave32 per edge; each lane owns D/32 consecutive features.
// Same-address loads of src/dst/dinv broadcast in L0.
// ---------------------------------------------------------------------------
__global__ void gcn_edge_agg_kernel(const int* __restrict__ src,
                                    const int* __restrict__ dst,
                                    const float* __restrict__ dinv,
                                    const float* __restrict__ h,
                                    float* __restrict__ agg,
                                    int E, int D) {
    const int wave = (blockIdx.x * blockDim.x + threadIdx.x) >> 5;
    const int lane = threadIdx.x & 31;
    if (wave >= E) return;                                    // wave-uniform

    const int s = src[wave];
    const int d = dst[wave];
    const float coef = dinv[s] * dinv[d];

    const int per = D >> 5;                                   // 4 (D=128) or 2 (D=64)
    const float* __restrict__ hs = h + (long long)s * D + lane * per;
    float* __restrict__ ad = agg + (long long)d * D + lane * per;
#pragma unroll 4
    for (int j = 0; j < per; ++j)
        atomicAdd(ad + j, hs[j] * coef);
}

// ---------------------------------------------------------------------------
// Host-side orchestration
// ---------------------------------------------------------------------------
static inline int ceil_div(long long a, int b) { return (int)((a + b - 1) / b); }

extern "C" void kernel_launch(void* const* d_in, const int* in_sizes, int n_in,
                              void* d_out, int out_size, void* d_ws, size_t ws_size,
                              hipStream_t stream) {
    (void)n_in; (void)ws_size;

    const float* W1 = (const float*)d_in[4];
    const float* b1 = (const float*)d_in[5];
    const float* W2 = (const float*)d_in[6];
    const float* b2 = (const float*)d_in[7];

    const int D1 = in_sizes[5];            // 128
    const int K1 = in_sizes[4] / D1;       // 64
    const int D2 = in_sizes[7];            // 64
    const int K2 = in_sizes[6] / D2;       // 128

    const int N1 = in_sizes[0] / K1;       // 50000
    const int N2 = in_sizes[2] / K1;       // 50000
    const int Nmax = (N1 > N2) ? N1 : N2;

    // Workspace layout (floats)
    float* ws    = (float*)d_ws;
    float* dinvA = ws;
    float* dinvB = dinvA + Nmax;
    float* h     = dinvB + Nmax;
    float* agg   = h + (long long)Nmax * D1;
    float* xmid  = agg + (long long)Nmax * D1;

    const int TPB = 256;
    long long out_off = 0;

    for (int g = 0; g < 2; ++g) {
        const float* x  = (const float*)d_in[g * 2];
        const int*   ei = (const int*)d_in[g * 2 + 1];
        const int N = in_sizes[g * 2] / K1;
        const int E = in_sizes[g * 2 + 1] / 2;
        const int* src = ei;
        const int* dst = ei + E;
        float* dinv = (g == 0) ? dinvA : dinvB;

        // deg = 1 (self loop) + in-degree ;  dinv = rsqrt(deg)
        gcn_fill_kernel<<<ceil_div(N, TPB), TPB, 0, stream>>>(dinv, 1.0f, N);
        gcn_deg_accum_kernel<<<ceil_div(E, TPB), TPB, 0, stream>>>(dst, dinv, E);
        gcn_rsqrt_kernel<<<ceil_div(N, TPB), TPB, 0, stream>>>(dinv, N);

        // ---- Layer 1: x[N,K1] @ W1[K1,D1] ----
        {
            const int tiles = (N >> 4) * (D1 >> 4);
            gcn_gemm_wmma_kernel<<<ceil_div(tiles, TPB / 32), TPB, 0, stream>>>(
                x, W1, b1, dinv, h, agg, N, K1, D1);
            gcn_edge_agg_kernel<<<ceil_div((long long)E * 32, TPB), TPB, 0, stream>>>(
                src, dst, dinv, h, agg, E, D1);
            gcn_relu_kernel<<<ceil_div((long long)N * D1, TPB), TPB, 0, stream>>>(
                agg, xmid, N * D1);
        }

        // ---- Layer 2: xmid[N,K2] @ W2[K2,D2] ----
        {
            const int tiles = (N >> 4) * (D2 >> 4);
            gcn_gemm_wmma_kernel<<<ceil_div(tiles, TPB / 32), TPB, 0, stream>>>(
                xmid, W2, b2, dinv, h, agg, N, K2, D2);
            gcn_edge_agg_kernel<<<ceil_div((long long)E * 32, TPB), TPB, 0, stream>>>(
                src, dst, dinv, h, agg, E, D2);
            gcn_relu_kernel<<<ceil_div((long long)N * D2, TPB), TPB, 0, stream>>>(
                agg, (float*)d_out + out_off, N * D2);
        }
        out_off += (long long)N * D2;
    }
    (void)out_size;
}